// SafetyLayer_57595511439508
// MI455X (gfx1250) — compile-verified
//
#include <hip/hip_runtime.h>
#include <stdint.h>

// ---------------------------------------------------------------------------
// SafetyLayer (EV-charging QP projection), MI455X / gfx1250.
//
// Pure streaming problem: 64 MiB in + 64 MiB out at 23.3 TB/s => ~5.8 us
// floor; ~1.6 FLOP/byte so matrix engines are useless. We exploit the
// CDNA5-specific data-movement path instead: Tensor Data Mover (TDM) async
// Global->LDS tile staging, double-buffered, fenced with s_wait_tensorcnt,
// with per-column QP bounds precomputed once per block and hoisted into
// registers. 128-bit LDS reads + 128-bit global stores.
// ---------------------------------------------------------------------------

static constexpr int kD         = 512;              // cars per row
static constexpr int kThreads   = 256;              // 8 wave32s per block
static constexpr int kTileRows  = 8;                // rows per TDM tile
static constexpr int kBlockRows = 32;               // rows per block
static constexpr int kNTiles    = kBlockRows / kTileRows;   // 4 (double buffered)
static constexpr int kTileElems = kTileRows * kD;   // 4096 floats = 16 KiB

typedef unsigned int u32;
typedef u32 v4u __attribute__((ext_vector_type(4)));
typedef int  v8i __attribute__((ext_vector_type(8)));
typedef int  v4i __attribute__((ext_vector_type(4)));

// Low 32 bits of a generic pointer derived from a __shared__ object are the
// byte offset within the workgroup's LDS allocation (addrspacecast 3->0 keeps
// the offset in the low dword; the aperture lives in the high dword).
__device__ __forceinline__ u32 lds_byte_off(const void* p) {
  return (u32)(uintptr_t)p;
}

// Issue one TDM transfer: a 1-D tile of kTileElems f32 from global to LDS.
// Descriptor per CDNA5 ISA ch.10.8 (D# groups 0/1; groups 2/3 zero => <=2D).
// This toolchain's builtin takes 6 args (extra int32x8 group before cpol).
__device__ __forceinline__ void tdm_load_tile_f32(const float* gsrc, u32 lds_addr) {
  const uint64_t ga = (uint64_t)(uintptr_t)gsrc;
  v4u g0;
  g0.x = 1u;                                              // count=1, user desc
  g0.y = lds_addr;                                        // lds_addr[31:0]
  g0.z = (u32)(ga & 0xffffffffu);                         // global_addr[31:0]
  g0.w = ((u32)(ga >> 32) & 0x01ffffffu) | (2u << 30);    // gaddr[56:32] | type=2
  v8i g1;
  g1[0] = (int)(2u << 16);                                // data_size=2 (4 bytes)
  g1[1] = (int)(((u32)kTileElems & 0xffffu) << 16);       // tensor_dim0[15:0]
  g1[2] = (int)((((u32)kTileElems >> 16) & 0xffffu)       // tensor_dim0[31:16]
                | (1u << 16));                            // tensor_dim1 = 1
  g1[3] = (int)(((u32)kTileElems & 0xffffu) << 16);       // tile_dim0 = 4096
  g1[4] = 0;                                              // tile_dim1/2 unused (1-D)
  g1[5] = (int)kTileElems;                                // dim0_stride (unused, 1-D)
  g1[6] = 0;
  g1[7] = 0;
  const v4i z4 = {0, 0, 0, 0};
  const v8i z8 = {0, 0, 0, 0, 0, 0, 0, 0};
  __builtin_amdgcn_tensor_load_to_lds(g0, g1, z4, z4, z8, /*cpol=*/0);
}

// Per-element projection. (soc_t + y/eta_d - C)/kslope rewritten as a*y + b
// with a = 1/(eta_d*kslope) hoisted; b1/b2 precomputed per column.
__device__ __forceinline__ float safety_elem(float xv, float ymin, float ymax,
                                             float b1, float b2, float adlo,
                                             float mask, float a, float acmax) {
  float y  = fminf(fmaxf(xv, ymin), ymax);   // optimal net action
  float ay = a * y;
  float m  = fmaxf(fmaxf(y, 0.0f), ay + b1); // AC >= max(y,0), SOC1 <= U
  m = fminf(m, fminf(acmax, y - adlo));      // AC <= ac_max, AD >= ad_lo
  m = fminf(m, ay + b2);                     // SOC1 >= L
  m = fmaxf(m, 0.0f);
  return m * mask;                           // inactive car -> 0
}

__global__ __launch_bounds__(kThreads) void safety_layer_kernel(
    const float* __restrict__ x, const float* __restrict__ t_rem,
    const float* __restrict__ soc_t, const float* __restrict__ t_dis,
    const float* __restrict__ soc_dis, float* __restrict__ out) {
  __shared__ __align__(16) float sYmin[kD], sYmax[kD], sB1[kD], sB2[kD],
                                 sAdlo[kD], sMask[kD];
  __shared__ __align__(16) float sBuf[2][kTileElems];   // double buffer, 32 KiB

  const int  tid     = threadIdx.x;
  const long rowBase = (long)blockIdx.x * kBlockRows;
  const float* gx    = x + rowBase * kD;
  float*       gout  = out + rowBase * kD;
  const bool issuer  = (tid < 32);           // wave 0 drives the TDM pipeline

  // Kick DMA of tile 0 immediately; it overlaps the param precompute below.
  if (issuer) tdm_load_tile_f32(gx, lds_byte_off(&sBuf[0][0]));

  constexpr float acmax  = 6.6f / 40.0f;     // ALPHA_C / BCAP
  constexpr float admag  = 6.6f / 40.0f;     // ALPHA_D / BCAP
  constexpr float U      = 0.97f;            // FINAL_SOC
  constexpr float etac   = 0.95f, etad = 0.95f;
  constexpr float kslope = 1.0f / etad - etac;

  // ---- per-column QP bounds (once per block; reference does this per sample)
  for (int j = tid; j < kD; j += kThreads) {
    const float tr = t_rem[j], st = soc_t[j], td = t_dis[j], sd = soc_dis[j];
    const float adlo = (td > 0.0f) ? -fminf(admag, etad * sd) : 0.0f;
    float L = U - (tr - 1.0f) * 6.6f * etac / 40.0f;
    L = fminf(fmaxf(L, 0.0f), U);
    // y_max: reduce discharge first, then charge headroom
    const float s_hi   = st + etac * acmax;
    const float excess = fmaxf(s_hi - U, 0.0f);
    const float ad_red = fminf(excess * etad, -adlo);
    const float rem    = fmaxf(excess - ad_red / etad, 0.0f);
    const float ac_hi  = fmaxf(acmax - rem / etac, 0.0f);
    const float ymax   = ac_hi - ad_red;
    // y_min: minimum net action subject to SOC1 >= L
    const float s_lo    = st + adlo / etad;
    const float deficit = fmaxf(L - s_lo, 0.0f);
    const float ad_inc  = fminf(deficit * etad, -adlo);
    const float rem2    = fmaxf(deficit - ad_inc / etad, 0.0f);
    const float ac_lo   = fminf(rem2 / etac, acmax);
    const float ymin    = fminf(ac_lo + adlo + ad_inc, ymax);
    sYmin[j] = ymin;
    sYmax[j] = ymax;
    sB1[j]   = (st - U) / kslope;
    sB2[j]   = (st - L) / kslope;
    sAdlo[j] = adlo;
    sMask[j] = (tr > 0.0f) ? 1.0f : 0.0f;
  }
  __syncthreads();

  // Each thread touches the same 4 columns in every pass of every tile:
  // col = (4*tid) mod 512. Hoist its params into registers.
  const int col0 = (tid * 4) & (kD - 1);
  const float4 ymin4 = *(const float4*)&sYmin[col0];
  const float4 ymax4 = *(const float4*)&sYmax[col0];
  const float4 b1v   = *(const float4*)&sB1[col0];
  const float4 b2v   = *(const float4*)&sB2[col0];
  const float4 adlo4 = *(const float4*)&sAdlo[col0];
  const float4 mask4 = *(const float4*)&sMask[col0];
  const float  a     = 1.0f / (etad * kslope);

  // ---- TDM double-buffered streaming pipeline
  for (int t = 0; t < kNTiles; ++t) {
    if (issuer) __builtin_amdgcn_s_wait_tensorcnt(0);  // tile t is in LDS
    __syncthreads();  // publish tile t; everyone done reading tile t-1
    if (issuer && (t + 1 < kNTiles)) {
      tdm_load_tile_f32(gx + (long)(t + 1) * kTileElems,
                        lds_byte_off(&sBuf[(t + 1) & 1][0]));
    }
    const float* __restrict__ buf = sBuf[t & 1];
    float* __restrict__ o = gout + (long)t * kTileElems;
#pragma unroll
    for (int p = 0; p < kTileElems / (kThreads * 4); ++p) {
      const int e = p * (kThreads * 4) + tid * 4;     // e mod 512 == col0
      const float4 xv = *(const float4*)&buf[e];      // ds_load_b128
      float4 r;
      r.x = safety_elem(xv.x, ymin4.x, ymax4.x, b1v.x, b2v.x, adlo4.x, mask4.x, a, acmax);
      r.y = safety_elem(xv.y, ymin4.y, ymax4.y, b1v.y, b2v.y, adlo4.y, mask4.y, a, acmax);
      r.z = safety_elem(xv.z, ymin4.z, ymax4.z, b1v.z, b2v.z, adlo4.z, mask4.z, a, acmax);
      r.w = safety_elem(xv.w, ymin4.w, ymax4.w, b1v.w, b2v.w, adlo4.w, mask4.w, a, acmax);
      *(float4*)&o[e] = r;                            // global_store_b128
    }
  }
}

extern "C" void kernel_launch(void* const* d_in, const int* in_sizes, int n_in,
                              void* d_out, int out_size, void* d_ws, size_t ws_size,
                              hipStream_t stream) {
  const float* x       = (const float*)d_in[0];
  const float* t_rem   = (const float*)d_in[1];
  const float* soc_t   = (const float*)d_in[2];
  const float* t_dis   = (const float*)d_in[3];
  const float* soc_dis = (const float*)d_in[4];
  float* out = (float*)d_out;

  const int batch = in_sizes[0] / kD;        // 32768
  const int grid  = batch / kBlockRows;      // 1024 blocks x 256 threads
  safety_layer_kernel<<<grid, kThreads, 0, stream>>>(x, t_rem, soc_t, t_dis,
                                                     soc_dis, out);
}